// CoreRNNFW_31731218383284
// MI455X (gfx1250) — compile-verified
//
#include <hip/hip_runtime.h>

typedef float v2f __attribute__((ext_vector_type(2)));
typedef float v8f __attribute__((ext_vector_type(8)));

#define T_STEPS 24
#define B_TOT   128
#define DGI     256
#define DHI     512
#define S_IT    3
#define LAMW    0.95f
#define ETAW    0.5f
#define EPS_A   1e-6f
#define LN_EPS  1e-5f
#define EPS_N   1e-6f
#define BPG     16       // batch rows per workgroup (== WMMA M)
#define NTHREADS 512     // 16 wave32 -> 4 waves per SIMD32, one wave per batch row

// butterfly sum across a wave32
__device__ __forceinline__ float wave_sum(float v) {
#pragma unroll
  for (int m = 16; m >= 1; m >>= 1) v += __shfl_xor(v, m, 32);
  return v;
}

// Persistent recurrent kernel: one workgroup owns 16 batch rows for all 24 steps.
// Fast-weight matrix A is never materialized: rank-r history factorization
// (hist rows + decayed coefficients) keeps all recurrent state L2-resident.
__global__ __launch_bounds__(NTHREADS)
void corernn_fw_kernel(const float* __restrict__ z_seq,   // [T,B,DG]
                       const float* __restrict__ W_h,     // [DH,DH]
                       const float* __restrict__ W_g,     // [DH,DG]
                       const float* __restrict__ b_h,     // [DH]
                       const float* __restrict__ gamma,   // [DH]
                       const float* __restrict__ beta,    // [DH]
                       const float* __restrict__ alpha_fw,// [1]
                       float* __restrict__ out,           // [B,DH]
                       float* __restrict__ hist)          // ws: [B,T,DH]
{
  extern __shared__ float smem[];
  float* x    = smem;                        // [16][512] h / h_s
  float* zbuf = x    + BPG*DHI;              // [2][16][256] double-buffered z_t
  float* hb   = zbuf + 2*BPG*DGI;            // [16][512] h_base
  float* Ahs  = hb   + BPG*DHI;              // [16][512] Ah / u
  float* gbuf = Ahs  + BPG*DHI;              // [16][24]  per-tau dots
  float* cw   = gbuf + BPG*T_STEPS;          // [16][24]  decayed Hebbian coeffs
  float* scal = cw   + BPG*T_STEPS;          // [16][8]   per-batch scalars
  const unsigned ZBUF_BYTE_OFF = (unsigned)(BPG*DHI) * 4u;  // LDS byte offset of zbuf

  const int tid  = threadIdx.x;
  const int b0   = blockIdx.x * BPG;
  const int lane = tid & 31;
  const int wave = tid >> 5;                 // 0..15; also the batch this wave reduces

  // k = compute_k(alpha_fw[0])
  const float afw  = alpha_fw[0];
  const float kexp = (afw >= 0.f) ? (1.f + log1pf(expf(afw)))
                                  : (1.f / (1.f + log1pf(expf(-afw))));

  // init: h = 0, coefficients = 0; stage z_0 synchronously into zbuf[0]
  for (int i = tid; i < BPG*DHI;     i += NTHREADS) x[i]  = 0.f;
  for (int i = tid; i < BPG*T_STEPS; i += NTHREADS) cw[i] = 0.f;
  for (int i = tid; i < BPG*DGI;     i += NTHREADS) zbuf[i] = z_seq[(size_t)b0*DGI + i];
  __syncthreads();

  // WMMA f32 16x16x4 per-lane fragment geometry (ISA 7.12.2):
  // lanes 0-15 hold (K=k0,k0+1) of row m=lane; lanes 16-31 hold (k0+2,k0+3).
  const int mrow  = lane & 15;
  const int khalf = (lane >> 4) << 1;

  for (int t = 0; t < T_STEPS; ++t) {
    const int cur = t & 1, nxt = cur ^ 1;
    // zbuf[cur] was async-prefetched during the previous step: drain ASYNCcnt.
    if (t > 0) asm volatile("s_wait_asynccnt 0" ::: "memory");
    __syncthreads();

    // ---- 1) h_base = h@W_h^T + z@W_g^T + b_h  (f32 WMMA, K=768) ----
    // Each wave owns two 16-wide N-tiles; A-fragment (h row) loaded once, fed to both.
    {
      const int n0a = wave * 16;             // tiles 0..15
      const int n0b = (wave + 16) * 16;      // tiles 16..31
      const float* wh0 = W_h + (size_t)(n0a + mrow) * DHI;
      const float* wh1 = W_h + (size_t)(n0b + mrow) * DHI;
      const float* wg0 = W_g + (size_t)(n0a + mrow) * DGI;
      const float* wg1 = W_g + (size_t)(n0b + mrow) * DGI;
      const float* xrow = x + mrow * DHI;
      const float* zrow = zbuf + cur * (BPG*DGI) + mrow * DGI;
      v8f acc0 = {}, acc1 = {};
      for (int k0 = 0; k0 < DHI; k0 += 4) {
        v2f a   = *(const v2f*)(xrow + k0 + khalf);
        v2f bf0 = *(const v2f*)(wh0  + k0 + khalf);
        v2f bf1 = *(const v2f*)(wh1  + k0 + khalf);
        acc0 = __builtin_amdgcn_wmma_f32_16x16x4_f32(false, a, false, bf0,
                                                     (short)0, acc0, false, false);
        acc1 = __builtin_amdgcn_wmma_f32_16x16x4_f32(false, a, false, bf1,
                                                     (short)0, acc1, false, false);
      }
      for (int k0 = 0; k0 < DGI; k0 += 4) {
        v2f a   = *(const v2f*)(zrow + k0 + khalf);
        v2f bf0 = *(const v2f*)(wg0  + k0 + khalf);
        v2f bf1 = *(const v2f*)(wg1  + k0 + khalf);
        acc0 = __builtin_amdgcn_wmma_f32_16x16x4_f32(false, a, false, bf0,
                                                     (short)0, acc0, false, false);
        acc1 = __builtin_amdgcn_wmma_f32_16x16x4_f32(false, a, false, bf1,
                                                     (short)0, acc1, false, false);
      }
      // D layout: VGPR r -> M = r (lanes 0-15) / r+8 (lanes 16-31), N = n0 + (lane&15)
      const int Mb = (lane >> 4) * 8;
      const int Na = n0a + mrow, Nb = n0b + mrow;
      const float bva = b_h[Na], bvb = b_h[Nb];
#pragma unroll
      for (int r = 0; r < 8; ++r) {
        hb[(Mb + r) * DHI + Na] = acc0[r] + bva;
        hb[(Mb + r) * DHI + Nb] = acc1[r] + bvb;
      }
    }
    __syncthreads();

    // ---- 2) h_s = relu(h_base) into x ----
    for (int i = tid; i < BPG*DHI; i += NTHREADS) x[i] = fmaxf(hb[i], 0.f);
    __syncthreads();

    // ---- 3) async-prefetch z_{t+1} into zbuf[nxt] (ASYNCcnt-tracked) ----
    if (t + 1 < T_STEPS) {
      const float* znext = z_seq + ((size_t)(t+1) * B_TOT + b0) * DGI;
      for (int task = tid; task < (BPG*DGI)/4; task += NTHREADS) {
        unsigned loff = ZBUF_BYTE_OFF + (unsigned)(nxt*(BPG*DGI) + task*4) * 4u;
        const float* g = znext + task*4;
        asm volatile("global_load_async_to_lds_b128 %0, %1, off"
                     :: "v"(loff), "v"(g) : "memory");
      }
    }

    // ---- 4) S inner associative-memory iterations (rank-t factorized A) ----
    for (int s = 0; s < S_IT; ++s) {
      // 4a) g[b][tau] = hist[b][tau] . h_s[b]
      if (t > 0) {
        for (int task = tid; task < BPG*t; task += NTHREADS) {
          int b = task / t, tau = task - b*t;
          const float* hrow = hist + (((size_t)(b0+b))*T_STEPS + tau)*DHI;
          const float* hsr  = x + b*DHI;
          float s0=0.f,s1=0.f,s2=0.f,s3=0.f;
          for (int j = 0; j < DHI; j += 4) {
            s0 += hrow[j+0]*hsr[j+0]; s1 += hrow[j+1]*hsr[j+1];
            s2 += hrow[j+2]*hsr[j+2]; s3 += hrow[j+3]*hsr[j+3];
          }
          gbuf[b*T_STEPS + tau] = (s0+s1)+(s2+s3);
        }
      }
      __syncthreads();
      // 4b) Ah[b][i] = sum_tau cw*g*hist  (coalesced along i)
      for (int task = tid; task < BPG*DHI; task += NTHREADS) {
        int b = task >> 9, i = task & 511;
        const float* hcol = hist + ((size_t)(b0+b))*T_STEPS*DHI + i;
        float sacc = 0.f;
        for (int tau = 0; tau < t; ++tau)
          sacc += cw[b*T_STEPS+tau] * gbuf[b*T_STEPS+tau] * hcol[(size_t)tau*DHI];
        Ahs[task] = sacc;
      }
      __syncthreads();
      // 4c) wave-per-batch reductions: dot, |h_s|^2, |Ah|^2 (shfl butterflies)
      {
        const float* hsr = x   + wave*DHI;
        const float* ar  = Ahs + wave*DHI;
        float d=0.f, nh=0.f, na=0.f;
        for (int j = lane; j < DHI; j += 32) {
          float u = hsr[j], v = ar[j];
          d += u*v; nh += u*u; na += v*v;
        }
        d = wave_sum(d); nh = wave_sum(nh); na = wave_sum(na);
        if (lane == 0) {
          float n1 = fmaxf(sqrtf(nh), EPS_N);
          float n2 = fmaxf(sqrtf(na), EPS_N);
          float R  = fminf(fmaxf(d/(n1*n2), 0.f), 1.f);
          float al = 1.f - powf(1.f - R, kexp);
          scal[wave*8+0] = al;
          scal[wave*8+1] = 1.f - al*al;
        }
      }
      __syncthreads();
      // 4d) u = (1-al^2)*h_base + al*Ah  (in place into Ahs)
      for (int task = tid; task < BPG*DHI; task += NTHREADS) {
        int b = task >> 9;
        Ahs[task] = scal[b*8+1]*hb[task] + scal[b*8+0]*Ahs[task];
      }
      __syncthreads();
      // 4e) layernorm stats (wave-per-batch)
      {
        const float* ur = Ahs + wave*DHI;
        float s0=0.f, s1=0.f;
        for (int j = lane; j < DHI; j += 32) { float u=ur[j]; s0+=u; s1+=u*u; }
        s0 = wave_sum(s0); s1 = wave_sum(s1);
        if (lane == 0) {
          float mean = s0 * (1.f/DHI);
          float var  = s1 * (1.f/DHI) - mean*mean;
          scal[wave*8+2] = mean;
          scal[wave*8+3] = rsqrtf(var + LN_EPS);
        }
      }
      __syncthreads();
      // 4f) h_s = relu(LN(u)) back into x
      for (int task = tid; task < BPG*DHI; task += NTHREADS) {
        int b = task >> 9, j = task & 511;
        float u = (Ahs[task] - scal[b*8+2]) * scal[b*8+3];
        x[task] = fmaxf(u * gamma[j] + beta[j], 0.f);
      }
      __syncthreads();
    }

    // ---- 5) Hebbian fast-weight update (factorized; skipped at last step) ----
    if (t < T_STEPS-1) {
      {
        const float* hsr = x + wave*DHI;
        float nh = 0.f;
        for (int j = lane; j < DHI; j += 32) { float u = hsr[j]; nh += u*u; }
        nh = wave_sum(nh);
        if (lane == 0) scal[wave*8+4] = ETAW / (nh + EPS_A);
      }
      __syncthreads();
      for (int task = tid; task < BPG*(t+1); task += NTHREADS) {
        int b = task / (t+1), tau = task - b*(t+1);
        if (tau < t) cw[b*T_STEPS+tau] *= LAMW;
        else         cw[b*T_STEPS+t]   = scal[b*8+4];
      }
      for (int task = tid; task < BPG*DHI; task += NTHREADS) {
        int b = task >> 9, j = task & 511;
        hist[(((size_t)(b0+b))*T_STEPS + t)*DHI + j] = x[task];
      }
      __syncthreads();
    }
  }

  // ---- final h ----
  for (int task = tid; task < BPG*DHI; task += NTHREADS) {
    int b = task >> 9, j = task & 511;
    out[(size_t)(b0+b)*DHI + j] = x[task];
  }
}

extern "C" void kernel_launch(void* const* d_in, const int* in_sizes, int n_in,
                              void* d_out, int out_size, void* d_ws, size_t ws_size,
                              hipStream_t stream) {
  (void)in_sizes; (void)n_in; (void)out_size; (void)ws_size;
  const float* z_seq = (const float*)d_in[0];
  const float* W_h   = (const float*)d_in[1];
  const float* W_g   = (const float*)d_in[2];
  const float* b_h   = (const float*)d_in[3];
  const float* gamma = (const float*)d_in[4];
  const float* beta  = (const float*)d_in[5];
  const float* afw   = (const float*)d_in[6];
  float* out  = (float*)d_out;
  float* hist = (float*)d_ws;   // needs 128*24*512*4 = 6.29 MB (L2-resident)

  const size_t smem_floats = (size_t)BPG*DHI        // x
                           + (size_t)2*BPG*DGI      // zbuf (double-buffered)
                           + (size_t)BPG*DHI        // hb
                           + (size_t)BPG*DHI        // Ahs
                           + (size_t)BPG*T_STEPS    // gbuf
                           + (size_t)BPG*T_STEPS    // cw
                           + (size_t)BPG*8;         // scal
  const size_t smem_bytes = smem_floats * sizeof(float);  // ~132 KB (< 320 KB/WGP)

  corernn_fw_kernel<<<dim3(B_TOT/BPG), dim3(NTHREADS), smem_bytes, stream>>>(
      z_seq, W_h, W_g, b_h, gamma, beta, afw, out, hist);
}